// RandWarpAug_76527727280285
// MI455X (gfx1250) — compile-verified
//
#include <hip/hip_runtime.h>
#include <math.h>

typedef __attribute__((ext_vector_type(2))) float v2f;
typedef __attribute__((ext_vector_type(8))) float v8f;

#define NB   512
#define NC   12
#define TT   4096
#define TD   1024
#define KS   51
#define NTHR 256

#define UPGUARD_LO 32
#define UPGUARD_HI 64
#define GOFF       16

__global__ void __launch_bounds__(NTHR)
randwarp_kernel(const float* __restrict__ src,
                const float* __restrict__ flow_mag,
                const float* __restrict__ noise,
                const float* __restrict__ gk,
                float* __restrict__ out)
{
    // sm_upb: [guard 32][up 4096][guard 64]; guards stay zero so the WMMA
    // im2col fragment loads need no bounds checks (index range [-25, 4122]).
    __shared__ float sm_upb[UPGUARD_LO + TT + UPGUARD_HI];
    __shared__ float sm_smooth[TT];
    __shared__ float sm_vec[2 * TD];
    __shared__ float sm_gb[96];       // zeroed; taps at [GOFF, GOFF+KS)

    float* const sm_up = sm_upb + UPGUARD_LO;

    const int tid = threadIdx.x;
    const int b   = blockIdx.x;
    const float fm = flow_mag[0];
    const float* nrow = noise + (size_t)b * TT;

    // ---- init guards + taps (Toeplitz index range [-15,67] -> +GOFF) -------
    for (int i = tid; i < 96; i += NTHR) sm_gb[i] = 0.0f;
    if (tid < UPGUARD_LO) sm_upb[tid] = 0.0f;
    if (tid < UPGUARD_HI) sm_upb[UPGUARD_LO + TT + tid] = 0.0f;
    __syncthreads();     // sm_gb zeroed before writing taps into it
    for (int i = tid; i < KS; i += NTHR) sm_gb[GOFF + i] = gk[i];

    // ---- Phase A: downsample x4 + magnitude x0.25, then /128 for VecInt ----
    // src positions are 4i+1.5 -> 0.5*(x[4i+1]+x[4i+2]); net scale 0.25*0.5/128
    for (int i = tid; i < TD; i += NTHR) {
        float a = nrow[4 * i + 1];
        float c = nrow[4 * i + 2];
        sm_vec[i] = fm * (a + c) * (1.0f / 1024.0f);
    }
    __syncthreads();

    // ---- Phase B: VecInt — 7 scaling-and-squaring self-warps in LDS --------
    int cur = 0;
    for (int s = 0; s < 7; ++s) {
        const float* vc = sm_vec + cur * TD;
        float*       vn = sm_vec + (cur ^ 1) * TD;
        for (int t = tid; t < TD; t += NTHR) {
            float f  = vc[t];
            float nl = (float)t + f;
            float v  = 2.0f * (nl / (float)(TD - 1) - 0.5f);
            float ix = (v + 1.0f) * 0.5f;
            float iy = (v + 1.0f) * 0.5f * (float)(TD - 1);
            float x0 = floorf(ix); float fx = ix - x0; int x0i = (int)x0;
            float xw = (1.0f - fx) * (((x0i >= 0) && (x0i <= 1)) ? 1.0f : 0.0f)
                     + fx * (((x0i + 1 >= 0) && (x0i + 1 <= 1)) ? 1.0f : 0.0f);
            float y0 = floorf(iy); float fy = iy - y0;
            int y0i = (int)y0, y1i = y0i + 1;
            float vy0 = (y0i >= 0 && y0i < TD) ? 1.0f : 0.0f;
            float vy1 = (y1i >= 0 && y1i < TD) ? 1.0f : 0.0f;
            int y0c = min(max(y0i, 0), TD - 1);
            int y1c = min(max(y1i, 0), TD - 1);
            float g0 = vc[y0c], g1 = vc[y1c];
            vn[t] = f + xw * ((1.0f - fy) * vy0 * g0 + fy * vy1 * g1);
        }
        cur ^= 1;
        __syncthreads();
    }

    // ---- Phase C: upsample x4 with magnitude x4 ----------------------------
    {
        const float* vf = sm_vec + cur * TD;
        for (int i = tid; i < TT; i += NTHR) {
            float sp = 0.25f * (float)i - 0.375f;
            sp = fminf(fmaxf(sp, 0.0f), (float)(TD - 1));
            int   i0 = (int)floorf(sp);
            int   i1 = min(i0 + 1, TD - 1);
            float w  = sp - (float)i0;
            sm_up[i] = 4.0f * (vf[i0] * (1.0f - w) + vf[i1] * w);
        }
    }
    __syncthreads();

    // ---- Phase D: 51-tap conv via V_WMMA_F32_16X16X4_F32 -------------------
    // D[m][n] = smooth[t0+16m+n] = sum_j A[m][j]*Bm[j][n]
    //   A[m][j]  = up[t0+16m+j-25]          (guard-padded, no branches)
    //   Bm[j][n] = g[j-n], zero outside     (guard-padded Toeplitz)
    // K padded to 68 -> 17 WMMA; 16 tiles of 256 outputs over 8 wave32s.
    {
        const int lane  = __builtin_amdgcn_workitem_id_x() & 31;
        const int wv    = __builtin_amdgcn_readfirstlane(tid >> 5); // SGPR-uniform
        const int lhalf = lane & 15;
        const int hsel  = (lane >> 4) & 1;            // 0: lanes 0-15, 1: lanes 16-31
        const int jadj  = hsel * 2;                   // A/B col pair K0+2/K0+3 on hi half
#pragma unroll 2
        for (int rep = 0; rep < 2; ++rep) {
            const int tile = wv + 8 * rep;
            const int t0   = tile * 256;
            const int pbase = t0 + 16 * lhalf + jadj - 25;   // A idx for k0=0 (.x)
            const int dbase = jadj - lhalf + GOFF;           // B idx for k0=0 (.x)
            v8f acc = {};
#pragma unroll
            for (int k0 = 0; k0 < 68; k0 += 4) {
                v2f a, bf;
                a.x  = sm_up[pbase + k0];
                a.y  = sm_up[pbase + k0 + 1];
                bf.x = sm_gb[dbase + k0];
                bf.y = sm_gb[dbase + k0 + 1];
                acc = __builtin_amdgcn_wmma_f32_16x16x4_f32(
                    false, a, false, bf, (short)0, acc, false, false);
            }
            // D layout: VGPR r -> (M=r+8*hsel, N=lane%16)
            const int mbase = hsel * 8;
#pragma unroll
            for (int r = 0; r < 8; ++r)
                sm_smooth[t0 + 16 * (mbase + r) + lhalf] = acc[r];
        }
    }
    __syncthreads();

    // ---- Phase E: per-thread warp parameters into registers ----------------
    float w0[16], w1[16];
    int   ia[16], ib[16];
#pragma unroll
    for (int k = 0; k < 16; ++k) {
        int   t  = tid + NTHR * k;
        float f  = sm_smooth[t];
        float nl = (float)t + f;
        float v  = 2.0f * (nl / (float)(TT - 1) - 0.5f);
        float ix = (v + 1.0f) * 0.5f;
        float iy = (v + 1.0f) * 0.5f * (float)(TT - 1);
        float x0 = floorf(ix); float fx = ix - x0; int x0i = (int)x0;
        float xw = (1.0f - fx) * (((x0i >= 0) && (x0i <= 1)) ? 1.0f : 0.0f)
                 + fx * (((x0i + 1 >= 0) && (x0i + 1 <= 1)) ? 1.0f : 0.0f);
        float y0 = floorf(iy); float fy = iy - y0;
        int y0i = (int)y0, y1i = y0i + 1;
        float vy0 = (y0i >= 0 && y0i < TT) ? 1.0f : 0.0f;
        float vy1 = (y1i >= 0 && y1i < TT) ? 1.0f : 0.0f;
        ia[k] = min(max(y0i, 0), TT - 1);
        ib[k] = min(max(y1i, 0), TT - 1);
        w0[k] = xw * (1.0f - fy) * vy0;
        w1[k] = xw * fy * vy1;
    }
    __syncthreads();  // sm_up free -> reuse as row buffer

    // ---- Phase F: warp all 12 channels, source row staged through LDS ------
    const size_t base = (size_t)b * NC * TT;
    float* rowbuf = sm_up;
    for (int c = 0; c < NC; ++c) {
        const float* sr = src + base + (size_t)c * TT;
        // prefetch next channel row while we work on this one
        if (c + 1 < NC) {
            const char* nx = (const char*)(sr + TT);
            __builtin_prefetch(nx + tid * 64, 0, 1);
        }
        const float4* sr4 = (const float4*)sr;
        float4*       rb4 = (float4*)rowbuf;
#pragma unroll
        for (int k = 0; k < 4; ++k) rb4[tid + NTHR * k] = sr4[tid + NTHR * k];
        __syncthreads();
        float* orow = out + base + (size_t)c * TT;
#pragma unroll
        for (int k = 0; k < 16; ++k)
            orow[tid + NTHR * k] = w0[k] * rowbuf[ia[k]] + w1[k] * rowbuf[ib[k]];
        __syncthreads();
    }
}

extern "C" void kernel_launch(void* const* d_in, const int* in_sizes, int n_in,
                              void* d_out, int out_size, void* d_ws, size_t ws_size,
                              hipStream_t stream) {
    const float* src   = (const float*)d_in[0];
    const float* fmag  = (const float*)d_in[1];
    const float* noise = (const float*)d_in[2];
    const float* gk    = (const float*)d_in[3];
    float* out = (float*)d_out;
    (void)in_sizes; (void)n_in; (void)out_size; (void)d_ws; (void)ws_size;
    hipLaunchKernelGGL(randwarp_kernel, dim3(NB), dim3(NTHR), 0, stream,
                       src, fmag, noise, gk, out);
}